// MSE_Ez___chamfer_loss_x_28123445854662
// MI455X (gfx1250) — compile-verified
//
#include <hip/hip_runtime.h>
#include <hip/hip_bf16.h>

typedef float v2f __attribute__((ext_vector_type(2)));
typedef float v8f __attribute__((ext_vector_type(8)));

#define NPTS   8192               // points per batch in out1 / lab1
#define NBATCH 2
#define JT     4                  // 16-wide y (column) tiles per block
#define WAVES  4                  // waves per chamfer block (split over x range)
#define TILES_PER_BATCH (NPTS / 16)   // 512
#define ITERS  (TILES_PER_BATCH / WAVES)  // 128 (compile-time trip count)
#define ALPHA_CHAMFER 0.1f

// ---------------------------------------------------------------------------
// zero the two accumulators in workspace
// ---------------------------------------------------------------------------
__global__ void init_accum_kernel(float* accum) {
    if (threadIdx.x < 2) accum[threadIdx.x] = 0.0f;
}

// ---------------------------------------------------------------------------
// Pre-pack x into the exact WMMA A-operand layout (done ONCE, reused by all
// chamfer blocks).  For x tile of 16 points (16x4 f32 A matrix):
//   lanes 0-15 : (x0, x1)          (K0,K1 for row M=lane)
//   lanes 16-31: (x2, ||x||^2)     (K2,K3 for row M=lane-16)
// pack[tile*32 + lane] = that lane's 2 VGPRs as float2.
// ---------------------------------------------------------------------------
__global__ void pack_x_kernel(const float* __restrict__ x,  // [B*NPTS, 3]
                              v2f* __restrict__ pack) {
    int wid  = threadIdx.x >> 5;
    int lane = threadIdx.x & 31;
    int tile = blockIdx.x * (blockDim.x >> 5) + wid;
    int hi   = lane >> 4;
    int t    = lane & 15;

    const float* xp = x + ((size_t)tile * 16 + t) * 3;
    float a0, a1;
    if (!hi) { a0 = xp[0]; a1 = xp[1]; }
    else     { a0 = xp[2]; a1 = 0.0f; }
    float s  = hi ? (a0 * a0) : (a0 * a0 + a1 * a1);
    float xx = s + __shfl_xor(s, 16);              // full ||x||^2
    if (hi) a1 = xx;                               // K3 slot carries ||x||^2
    v2f a; a[0] = a0; a[1] = a1;
    pack[(size_t)tile * 32 + lane] = a;
}

// ---------------------------------------------------------------------------
// MSE: sum of (a-b)^2 over n float4s -> atomicAdd into accum[0]
// ---------------------------------------------------------------------------
__global__ void mse_kernel(const float4* __restrict__ a,
                           const float4* __restrict__ b,
                           float* __restrict__ accum, int n4) {
    float s = 0.0f;
    for (int i = blockIdx.x * blockDim.x + threadIdx.x; i < n4;
         i += gridDim.x * blockDim.x) {
        float4 va = a[i], vb = b[i];
        float d0 = va.x - vb.x, d1 = va.y - vb.y;
        float d2 = va.z - vb.z, d3 = va.w - vb.w;
        s += d0 * d0 + d1 * d1 + d2 * d2 + d3 * d3;
    }
    #pragma unroll
    for (int off = 16; off > 0; off >>= 1) s += __shfl_xor(s, off);

    __shared__ float wsum[8];
    int lane = threadIdx.x & 31;
    int wid  = threadIdx.x >> 5;
    if (lane == 0) wsum[wid] = s;
    __syncthreads();
    if (threadIdx.x == 0) {
        float t = 0.0f;
        int nw = blockDim.x >> 5;
        for (int i = 0; i < nw; ++i) t += wsum[i];
        atomicAdd(accum, t);
    }
}

// ---------------------------------------------------------------------------
// Chamfer: for each y_j, min_i ||x_i - y_j||^2, summed into accum[1].
//
// D[i,j] = A.B with A row i = (x0,x1,x2,||x_i||^2), B col j = (-2y0,-2y1,-2y2,1)
//        = ||x_i||^2 - 2 x.y ;  ||y_j||^2 added once after the min loop.
//
// Block = 4 waves; each wave covers a strided quarter of the 512 x tiles.
// Hot loop is software-pipelined: tile k+1's global_load_b64 is issued before
// tile k's 4 WMMAs + 16 min3s, so the loadcnt wait overlaps the matrix work.
// ---------------------------------------------------------------------------
__global__ void chamfer_kernel(const v2f* __restrict__ pack, // packed A tiles
                               const float* __restrict__ y,  // [B, NPTS, 3]
                               float* __restrict__ accum) {
    const int jblocks = NPTS / (16 * JT);   // 128
    int b    = blockIdx.x / jblocks;
    int jblk = blockIdx.x % jblocks;
    int wid  = threadIdx.x >> 5;
    int lane = threadIdx.x & 31;
    int hi   = lane >> 4;
    int t    = lane & 15;

    // ---- B operands (y tiles) and ||y||^2 (one-time setup) ----
    const float* yb = y + (size_t)b * NPTS * 3;
    v2f   bm[JT];
    float yy[JT];
    #pragma unroll
    for (int q = 0; q < JT; ++q) {
        const float* yp = yb + (size_t)((jblk * JT + q) * 16 + t) * 3;
        float b0, b1;
        if (!hi) { b0 = -2.0f * yp[0]; b1 = -2.0f * yp[1]; }
        else     { b0 = -2.0f * yp[2]; b1 = 1.0f; }
        bm[q][0] = b0; bm[q][1] = b1;
        float s = hi ? (b0 * b0) : (b0 * b0 + b1 * b1);
        yy[q] = 0.25f * (s + __shfl_xor(s, 16));   // ||y_j||^2
    }

    float mn[JT];
    #pragma unroll
    for (int q = 0; q < JT; ++q) mn[q] = 3.0e38f;

    // ---- hot loop: prefetch next A tile, then 4 WMMAs + row-mins ----
    const v2f* xp = pack + ((size_t)b * TILES_PER_BATCH + wid) * 32 + lane;
    const int STRIDE = WAVES * 32;                 // v2f elements per k-step

    v2f a = xp[0];                                 // prologue load
    for (int k = 0; k < ITERS - 1; ++k) {
        v2f an = xp[(size_t)(k + 1) * STRIDE];     // prefetch tile k+1
        #pragma unroll
        for (int q = 0; q < JT; ++q) {
            v8f c = {};
            c = __builtin_amdgcn_wmma_f32_16x16x4_f32(
                    false, a, false, bm[q], (short)0, c, false, false);
            float m = mn[q];
            #pragma unroll
            for (int r = 0; r < 8; ++r) m = fminf(m, c[r]);
            mn[q] = m;
        }
        a = an;                                    // wait lands here, post-WMMA
    }
    // epilogue: last tile
    #pragma unroll
    for (int q = 0; q < JT; ++q) {
        v8f c = {};
        c = __builtin_amdgcn_wmma_f32_16x16x4_f32(
                false, a, false, bm[q], (short)0, c, false, false);
        float m = mn[q];
        #pragma unroll
        for (int r = 0; r < 8; ++r) m = fminf(m, c[r]);
        mn[q] = m;
    }

    // ---- combine the 4 waves' partial mins via LDS ----
    __shared__ float lmin[WAVES][JT][32];
    #pragma unroll
    for (int q = 0; q < JT; ++q) lmin[wid][q][lane] = mn[q];
    __syncthreads();

    if (wid == 0) {
        float total = 0.0f;
        #pragma unroll
        for (int q = 0; q < JT; ++q) {
            float m = lmin[0][q][lane];
            #pragma unroll
            for (int w = 1; w < WAVES; ++w) m = fminf(m, lmin[w][q][lane]);
            m = fminf(m, __shfl_xor(m, 16));       // other 8 rows
            total += m + yy[q];                    // duplicated on both halves
        }
        #pragma unroll
        for (int off = 16; off > 0; off >>= 1) total += __shfl_xor(total, off);
        if (lane == 0) atomicAdd(accum + 1, 0.5f * total); // halves counted 2x
    }
}

// ---------------------------------------------------------------------------
// finalize: (loss, ez, rp, -1)
// ---------------------------------------------------------------------------
__global__ void finalize_kernel(const float* __restrict__ accum,
                                float* __restrict__ out) {
    float ez = accum[0] * (1.0f / (float)(NBATCH * 2 * 512 * 512));
    float rp = accum[1] * (1.0f / (float)(NBATCH * NPTS));
    out[0] = ez + ALPHA_CHAMFER * rp;
    out[1] = ez;
    out[2] = rp;
    out[3] = -1.0f;
}

// ---------------------------------------------------------------------------
extern "C" void kernel_launch(void* const* d_in, const int* in_sizes, int n_in,
                              void* d_out, int out_size, void* d_ws, size_t ws_size,
                              hipStream_t stream) {
    const float* out0 = (const float*)d_in[0];   // [2,2,512,512]
    const float* out1 = (const float*)d_in[1];   // [2,8192,3]
    const float* lab0 = (const float*)d_in[2];   // [2,2,512,512]
    const float* lab1 = (const float*)d_in[3];   // [2,8192,3]
    float* out   = (float*)d_out;

    float* accum = (float*)d_ws;                         // 2 floats
    v2f*   pack  = (v2f*)((char*)d_ws + 256);            // 256 KB packed A tiles

    init_accum_kernel<<<1, 32, 0, stream>>>(accum);

    // pack all x tiles: NBATCH*512 = 1024 tiles, 8 waves/block -> 128 blocks
    pack_x_kernel<<<(NBATCH * TILES_PER_BATCH) / 8, 256, 0, stream>>>(out1, pack);

    int n4 = in_sizes[0] / 4;                            // 262144 float4s
    mse_kernel<<<512, 256, 0, stream>>>((const float4*)out0,
                                        (const float4*)lab0, accum, n4);

    int nblocks = NBATCH * (NPTS / (16 * JT));           // 256 blocks, 4 waves each
    chamfer_kernel<<<nblocks, 32 * WAVES, 0, stream>>>(pack, lab1, accum);

    finalize_kernel<<<1, 1, 0, stream>>>(accum, out);
}